// MaxPool_83339545412231
// MI455X (gfx1250) — compile-verified
//
#include <hip/hip_runtime.h>

// Graph max-pool: msgs = h[src]; h_max = segment_max(msgs, dst); out = h_max[pool]
// (nodes with no in-edges -> 0).
//
// Strategy (MI455X): everything (h 12.8MB + scratch 12.8MB) is L2-resident
// (192MB L2), so the bottleneck is 51.2M scatter-max RMWs. Use native
// single-instruction no-return global_atomic_max_u32 on a monotonic unsigned
// key encoding of float, 16 lanes/edge with 128-bit row-segment loads.

#define D_FEAT 64

// Monotonic order-preserving float<->uint key.
//  f >= 0 : key = bits | 0x80000000  (>= 0x80000000)
//  f <  0 : key = ~bits              (<  0x7FFFFFFF for finite negatives)
// key == 0 is unreachable from finite floats -> used as "untouched" marker.
__device__ __forceinline__ unsigned f2key(float f) {
    unsigned b = __float_as_uint(f);
    return (b & 0x80000000u) ? ~b : (b | 0x80000000u);
}
__device__ __forceinline__ float key2f(unsigned k) {
    unsigned b = (k & 0x80000000u) ? (k ^ 0x80000000u) : ~k;
    return __uint_as_float(b);
}

__global__ void gmp_init_ws(unsigned* __restrict__ ws, int n4) {
    int i = blockIdx.x * blockDim.x + threadIdx.x;
    if (i < n4) {
        ((uint4*)ws)[i] = make_uint4(0u, 0u, 0u, 0u);
    }
}

__global__ void gmp_edge_scatter_max(const float* __restrict__ h,
                                     const int* __restrict__ src,
                                     const int* __restrict__ dst,
                                     unsigned* __restrict__ ws,
                                     int n_edges) {
    long tid = (long)blockIdx.x * blockDim.x + threadIdx.x;
    int e = (int)(tid >> 4);
    if (e >= n_edges) return;
    int l = ((int)tid & 15) << 2;  // feature offset: 0,4,...,60

    int s = src[e];
    int d = dst[e];

    // 128-bit coalesced gather of one 4-float segment (1/16 row) of the source node.
    const float4 v = *(const float4*)(h + (long)s * D_FEAT + l);

    unsigned* w = ws + (long)d * D_FEAT + l;
    // No-return u32 max atomics -> single global_atomic_max_u32 each,
    // tracked on STOREcnt (wave does not wait for a returned value).
    atomicMax(w + 0, f2key(v.x));
    atomicMax(w + 1, f2key(v.y));
    atomicMax(w + 2, f2key(v.z));
    atomicMax(w + 3, f2key(v.w));
}

__global__ void gmp_pool_gather(const unsigned* __restrict__ ws,
                                const int* __restrict__ pool,
                                float* __restrict__ out,
                                int n_pool) {
    long tid = (long)blockIdx.x * blockDim.x + threadIdx.x;
    int p = (int)(tid >> 4);
    if (p >= n_pool) return;
    int l = ((int)tid & 15) << 2;

    int node = pool[p];
    uint4 k = *(const uint4*)(ws + (long)node * D_FEAT + l);

    float4 o;
    o.x = k.x ? key2f(k.x) : 0.0f;   // key==0 => no in-edges => 0
    o.y = k.y ? key2f(k.y) : 0.0f;
    o.z = k.z ? key2f(k.z) : 0.0f;
    o.w = k.w ? key2f(k.w) : 0.0f;

    *(float4*)(out + (long)p * D_FEAT + l) = o;
}

extern "C" void kernel_launch(void* const* d_in, const int* in_sizes, int n_in,
                              void* d_out, int out_size, void* d_ws, size_t ws_size,
                              hipStream_t stream) {
    const float* h    = (const float*)d_in[0];
    const int*   src  = (const int*)d_in[1];
    const int*   dst  = (const int*)d_in[2];
    const int*   pool = (const int*)d_in[3];
    float*       out  = (float*)d_out;
    unsigned*    ws   = (unsigned*)d_ws;

    const int n_nodes = in_sizes[0] / D_FEAT;
    const int n_edges = in_sizes[1];
    const int n_pool  = in_sizes[3];

    const int block = 256;  // 8 wave32s

    // 1) Reset scratch every call (harness does not re-poison between replays).
    int n4 = (n_nodes * D_FEAT) / 4;
    gmp_init_ws<<<(n4 + block - 1) / block, block, 0, stream>>>(ws, n4);

    // 2) Scatter-max over edges: 16 lanes per edge.
    long ethreads = (long)n_edges * 16;
    gmp_edge_scatter_max<<<(unsigned)((ethreads + block - 1) / block), block, 0, stream>>>(
        h, src, dst, ws, n_edges);

    // 3) Decode + pooling gather: 16 lanes per pooled node.
    long pthreads = (long)n_pool * 16;
    gmp_pool_gather<<<(unsigned)((pthreads + block - 1) / block), block, 0, stream>>>(
        ws, pool, out, n_pool);
}